// Model_29377576304759
// MI455X (gfx1250) — compile-verified
//
#include <hip/hip_runtime.h>
#include <math.h>

typedef __attribute__((ext_vector_type(2))) float v2f;
typedef __attribute__((ext_vector_type(8))) float v8f;
typedef __attribute__((ext_vector_type(4))) unsigned u32x4;
typedef __attribute__((ext_vector_type(8))) int i32x8;
typedef __attribute__((ext_vector_type(4))) int i32x4;

#define BN_EPS 1e-5f
#define HID 256
#define KDIM 4096
#define CHUNK 512          // K-chunk staged per TDM transfer (16 rows x 512 f32 = 32 KB)
#define NCHUNK (KDIM / CHUNK)

__device__ __forceinline__ float sigmoidf_(float x) {
    return 1.f / (1.f + __expf(-x));
}

// ---------------------------------------------------------------------------
// Implicit-GEMM conv3x3 (pad=1) + BN + ReLU, fp32 WMMA 16x16x4.
// One wave computes a 16(oc) x 16(pixel) output tile.
//   A (16x4)  : weights, lane l16 holds row M=l16, K = k0 + 2*half (+1)
//   B (4x16)  : im2col patch, lane l16 holds col N=l16 (pixel), same K split
//   C/D (16x16): acc[r] = element (M = r + 8*half, N = l16)
// ---------------------------------------------------------------------------
template<int CIN, int COUT, int STRIDE, int HIN, int WIN>
__global__ __launch_bounds__(256) void conv_bn_relu_wmma(
    const float* __restrict__ x, const float* __restrict__ w,
    const float* __restrict__ gam, const float* __restrict__ bet,
    const float* __restrict__ mu,  const float* __restrict__ var,
    float* __restrict__ y, int nimg)
{
    constexpr int HOUT = HIN / STRIDE;
    constexpr int WOUT = WIN / STRIDE;
    constexpr int NPIX = HOUT * WOUT;
    constexpr int PIX_TILES = NPIX / 16;
    constexpr int OC_TILES  = COUT / 16;
    constexpr int K = CIN * 9;

    const int lane = threadIdx.x & 31;
    const int wave = threadIdx.x >> 5;
    const int half = lane >> 4;     // 0: K pair {0,1}, 1: K pair {2,3}
    const int l16  = lane & 15;

    int tile  = blockIdx.x * (blockDim.x >> 5) + wave;  // wave-uniform
    int total = nimg * OC_TILES * PIX_TILES;
    if (tile >= total) return;

    int img = tile / (OC_TILES * PIX_TILES);
    int rem = tile - img * (OC_TILES * PIX_TILES);
    int oct = rem / PIX_TILES;
    int pt  = rem - oct * PIX_TILES;

    const float* wrow = w + (size_t)(oct * 16 + l16) * K;
    int pix = pt * 16 + l16;
    int oh = pix / WOUT, ow = pix - oh * WOUT;
    const float* xin = x + (size_t)img * CIN * HIN * WIN;

    v8f acc = {};
    for (int k0 = 0; k0 < K; k0 += 4) {
        int ka = k0 + 2 * half;
        v2f a;
        a.x = (ka     < K) ? wrow[ka]     : 0.f;
        a.y = (ka + 1 < K) ? wrow[ka + 1] : 0.f;
        float bv[2];
#pragma unroll
        for (int t = 0; t < 2; ++t) {
            int k = ka + t;
            float v = 0.f;
            if (k < K) {
                int ic = k / 9;
                int r9 = k - ic * 9;
                int kh = r9 / 3, kw = r9 - kh * 3;
                int ih = oh * STRIDE + kh - 1;   // pad = 1
                int iw = ow * STRIDE + kw - 1;
                if (ih >= 0 && ih < HIN && iw >= 0 && iw < WIN)
                    v = xin[(size_t)(ic * HIN + ih) * WIN + iw];
            }
            bv[t] = v;
        }
        v2f b; b.x = bv[0]; b.y = bv[1];
        acc = __builtin_amdgcn_wmma_f32_16x16x4_f32(
            false, a, false, b, (short)0, acc, false, false);
    }

#pragma unroll
    for (int r = 0; r < 8; ++r) {
        int oc = oct * 16 + r + 8 * half;
        float inv  = gam[oc] * rsqrtf(var[oc] + BN_EPS);
        float beta = bet[oc] - mu[oc] * inv;
        float val  = acc[r] * inv + beta;
        y[((size_t)img * COUT + oc) * NPIX + pix] = fmaxf(val, 0.f);
    }
}

// ---------------------------------------------------------------------------
// TDM helper: DMA a 2D tile (16 rows x CHUNK f32, row stride KDIM f32) from
// global memory into LDS at byte offset lds_addr. D# layout per CDNA5 ISA §8:
//   group0: count=1 | lds_addr | global_addr | type=2
//   group1: data_size=4B, tensor_dim0=KDIM, tensor_dim1=1024,
//           tile_dim0=CHUNK, tile_dim1=16, tensor_dim0_stride=KDIM
//   groups 2/3 (+ spare group): zero (2D tensor)
// Toolchain here is the 6-arg clang-23 form:
//   (u32x4 g0, i32x8 g1, i32x4 g2, i32x4 g3, i32x8 spare, i32 cpol)
// ---------------------------------------------------------------------------
__device__ __forceinline__ void tdm_load_tile16xCHUNK(unsigned lds_addr,
                                                      const float* gsrc)
{
    unsigned long long ga = (unsigned long long)(uintptr_t)gsrc;
    u32x4 g0;
    g0[0] = 1u;                                              // count = 1
    g0[1] = lds_addr;                                        // LDS byte address
    g0[2] = (unsigned)(ga & 0xffffffffu);                    // global addr lo
    g0[3] = (unsigned)((ga >> 32) & 0x01ffffffu) | (2u << 30); // addr hi | type=2
    i32x8 g1;
    g1[0] = (int)(2u << 16);                 // data_size = 2 (4 bytes)
    g1[1] = (int)((unsigned)KDIM << 16);     // tensor_dim0 lo16 in [63:48]
    g1[2] = (int)(1024u << 16);              // tensor_dim0 hi=0 | tensor_dim1 lo16
    g1[3] = (int)((unsigned)CHUNK << 16);    // tensor_dim1 hi=0 | tile_dim0
    g1[4] = 16;                              // tile_dim1 = 16 rows (tile_dim2=0)
    g1[5] = KDIM;                            // tensor_dim0_stride lo32
    g1[6] = 0;                               // stride hi | dim1_stride lo
    g1[7] = 0;
    i32x4 z4 = {0, 0, 0, 0};
    i32x8 z8 = {0, 0, 0, 0, 0, 0, 0, 0};
    __builtin_amdgcn_tensor_load_to_lds(g0, g1, z4, z4, z8, 0);
}

// ---------------------------------------------------------------------------
// Xg[img][gate] = feats[img] . w_ih[gate] + b_ih[gate] + b_hh[gate]
// M = 1024 gates, N = 1024 images (b*64+s), K = 4096.
// Block = 8 waves: one shared image tile (B staged in LDS by the Tensor Data
// Mover, double-buffered) x 8 gate tiles. A streams from global (L2-resident).
// ---------------------------------------------------------------------------
__global__ __launch_bounds__(256) void xgemm_wmma(
    const float* __restrict__ feats, const float* __restrict__ w_ih,
    const float* __restrict__ b_ih,  const float* __restrict__ b_hh,
    float* __restrict__ xg)
{
    __shared__ float bsh[2][16 * CHUNK];   // 64 KB double buffer

    const int lane = threadIdx.x & 31;
    const int wave = threadIdx.x >> 5;
    const int half = lane >> 4;
    const int l16  = lane & 15;

    const int it = blockIdx.x >> 3;                   // image tile 0..63
    const int gt = ((blockIdx.x & 7) << 3) + wave;    // gate tile 0..63

    const float* arow  = w_ih  + (size_t)(gt * 16 + l16) * KDIM;
    const float* gtile = feats + (size_t)it * 16 * KDIM;   // B tile start

    const unsigned lds_base =
        (unsigned)(uintptr_t)(__attribute__((address_space(3))) void*)&bsh[0][0];
    const unsigned buf_bytes = 16 * CHUNK * 4;

    if (wave == 0) {
        tdm_load_tile16xCHUNK(lds_base, gtile);             // chunk 0 -> buf 0
        __builtin_amdgcn_s_wait_tensorcnt(0);
    }
    __syncthreads();

    v8f acc = {};
    for (int ci = 0; ci < NCHUNK; ++ci) {
        if (wave == 0 && ci + 1 < NCHUNK)                   // overlap DMA
            tdm_load_tile16xCHUNK(lds_base + ((ci + 1) & 1) * buf_bytes,
                                  gtile + (ci + 1) * CHUNK);
        if (ci + 1 < NCHUNK)
            __builtin_prefetch(arow + (ci + 1) * CHUNK, 0, 0);  // global_prefetch_b8

        const float* bbuf = &bsh[ci & 1][0];
        const float* ac   = arow + ci * CHUNK;
#pragma unroll 8
        for (int k0 = 0; k0 < CHUNK; k0 += 4) {
            int ka = k0 + 2 * half;                 // even -> 8B aligned
            v2f a = *(const v2f*)&ac[ka];
            v2f b = *(const v2f*)&bbuf[l16 * CHUNK + ka];   // ds_load_b64
            acc = __builtin_amdgcn_wmma_f32_16x16x4_f32(
                false, a, false, b, (short)0, acc, false, false);
        }
        if (wave == 0 && ci + 1 < NCHUNK)
            __builtin_amdgcn_s_wait_tensorcnt(0);
        __syncthreads();
    }

#pragma unroll
    for (int r = 0; r < 8; ++r) {
        int gate = gt * 16 + r + 8 * half;
        int img  = it * 16 + l16;
        xg[(size_t)img * 1024 + gate] = acc[r] + b_ih[gate] + b_hh[gate];
    }
}

// ---------------------------------------------------------------------------
// Persistent single-block LSTM + classifier head.
// h,c (16x256) and gates (1024x16) in LDS (96 KB, CDNA5 WGP has 320 KB).
// ---------------------------------------------------------------------------
__global__ __launch_bounds__(1024) void lstm_head_wmma(
    const float* __restrict__ xg, const float* __restrict__ w_hh,
    const float* __restrict__ cls_w, const float* __restrict__ cls_b,
    float* __restrict__ out)
{
    __shared__ float h_lds[16 * HID];     // [batch][hidden]
    __shared__ float c_lds[16 * HID];
    __shared__ float g_lds[16 * 1024];    // [gate][batch]

    const int tid  = threadIdx.x;
    const int lane = tid & 31;
    const int wave = tid >> 5;
    const int half = lane >> 4;
    const int l16  = lane & 15;

    for (int i = tid; i < 16 * HID; i += 1024) { h_lds[i] = 0.f; c_lds[i] = 0.f; }
    __syncthreads();

    for (int s = 0; s < 64; ++s) {
        // gates = Xg(s) + h @ w_hh^T   (Xg already includes b_ih + b_hh)
        for (int gt = wave; gt < 64; gt += 32) {       // wave-uniform loop
            v8f acc;
#pragma unroll
            for (int r = 0; r < 8; ++r)
                acc[r] = xg[(size_t)(l16 * 64 + s) * 1024 + (gt * 16 + r + 8 * half)];

            const float* arow = w_hh + (size_t)(gt * 16 + l16) * HID;
            for (int k0 = 0; k0 < HID; k0 += 4) {
                int ka = k0 + 2 * half;
                v2f a = *(const v2f*)&arow[ka];
                v2f b = *(const v2f*)&h_lds[l16 * HID + ka];
                acc = __builtin_amdgcn_wmma_f32_16x16x4_f32(
                    false, a, false, b, (short)0, acc, false, false);
            }
#pragma unroll
            for (int r = 0; r < 8; ++r)
                g_lds[(gt * 16 + r + 8 * half) * 16 + l16] = acc[r];
        }
        __syncthreads();

        // cell update
        for (int i = tid; i < 16 * HID; i += 1024) {
            int b  = i / HID;
            int hu = i - b * HID;
            float ig = sigmoidf_(g_lds[(      hu) * 16 + b]);
            float fg = sigmoidf_(g_lds[(256 + hu) * 16 + b]);
            float gg = tanhf    (g_lds[(512 + hu) * 16 + b]);
            float og = sigmoidf_(g_lds[(768 + hu) * 16 + b]);
            float cn = fg * c_lds[i] + ig * gg;
            c_lds[i] = cn;
            h_lds[i] = og * tanhf(cn);
        }
        __syncthreads();

        // classifier head: out[b][s-11] = tanh(h_b . cls_w + cls_b), s >= 11
        if (s >= 11 && tid < 16) {
            float acc = cls_b[0];
            for (int k = 0; k < HID; ++k)
                acc += h_lds[tid * HID + k] * cls_w[k];
            out[tid * 53 + (s - 11)] = tanhf(acc);
        }
        __syncthreads();
    }
}

// ---------------------------------------------------------------------------
extern "C" void kernel_launch(void* const* d_in, const int* in_sizes, int n_in,
                              void* d_out, int out_size, void* d_ws, size_t ws_size,
                              hipStream_t stream)
{
    const float* imgs = (const float*)d_in[0];
    const float *W[6], *G[6], *Bb[6], *M[6], *V[6];
    for (int i = 0; i < 6; ++i) {
        W[i]  = (const float*)d_in[1 + 5 * i];
        G[i]  = (const float*)d_in[2 + 5 * i];
        Bb[i] = (const float*)d_in[3 + 5 * i];
        M[i]  = (const float*)d_in[4 + 5 * i];
        V[i]  = (const float*)d_in[5 + 5 * i];
    }
    const float* w_ih  = (const float*)d_in[31];
    const float* w_hh  = (const float*)d_in[32];
    const float* b_ih  = (const float*)d_in[33];
    const float* b_hh  = (const float*)d_in[34];
    const float* cls_w = (const float*)d_in[35];
    const float* cls_b = (const float*)d_in[36];
    float* out = (float*)d_out;

    char*  ws   = (char*)d_ws;
    float* bufA = (float*)ws;                          // 64 MB
    float* bufB = (float*)(ws + ((size_t)64 << 20));   // 64 MB
    const int NIMG = 16 * 64;

    // conv stack, ping-pong A/B. tiles = nimg * (Cout/16) * (Hout*Wout/16)
    conv_bn_relu_wmma<3, 16, 2, 64, 64><<< (NIMG * 1 * 64 + 7) / 8, 256, 0, stream >>>(
        imgs, W[0], G[0], Bb[0], M[0], V[0], bufA, NIMG);
    conv_bn_relu_wmma<16, 16, 1, 32, 32><<< (NIMG * 1 * 64 + 7) / 8, 256, 0, stream >>>(
        bufA, W[1], G[1], Bb[1], M[1], V[1], bufB, NIMG);
    conv_bn_relu_wmma<16, 32, 2, 32, 32><<< (NIMG * 2 * 16 + 7) / 8, 256, 0, stream >>>(
        bufB, W[2], G[2], Bb[2], M[2], V[2], bufA, NIMG);
    conv_bn_relu_wmma<32, 32, 1, 16, 16><<< (NIMG * 2 * 16 + 7) / 8, 256, 0, stream >>>(
        bufA, W[3], G[3], Bb[3], M[3], V[3], bufB, NIMG);
    conv_bn_relu_wmma<32, 64, 2, 16, 16><<< (NIMG * 4 * 4 + 7) / 8, 256, 0, stream >>>(
        bufB, W[4], G[4], Bb[4], M[4], V[4], bufA, NIMG);
    conv_bn_relu_wmma<64, 64, 1, 8, 8><<< (NIMG * 4 * 4 + 7) / 8, 256, 0, stream >>>(
        bufA, W[5], G[5], Bb[5], M[5], V[5], bufB, NIMG);

    // LSTM input projection: feats (bufB) -> Xg (bufA, 4 MB), TDM-staged B
    xgemm_wmma<<< 512, 256, 0, stream >>>(bufB, w_ih, b_ih, b_hh, bufA);

    // recurrence + head
    lstm_head_wmma<<< 1, 1024, 0, stream >>>(bufA, w_hh, cls_w, cls_b, out);
}